// WordEmbedding_43181601194151
// MI455X (gfx1250) — compile-verified
//
#include <hip/hip_runtime.h>

// WordEmbedding gather: out[b,s,:] = weight[ids[b,s], :]
// weight: [50257, 1024] fp32, ids: [16, 2048] int, out: [16*2048, 1024] fp32.
// Pure memory-bound row gather; target = HBM bandwidth (23.3 TB/s -> ~11us).

typedef __attribute__((ext_vector_type(4))) float float4v;

static constexpr int DIM              = 1024;       // embedding dim
static constexpr int VEC              = 4;          // floats per thread per row
static constexpr int THREADS          = DIM / VEC;  // 256 threads = one 4KiB row
static constexpr int TOKENS_PER_BLOCK = 4;          // 4 B128 loads in flight/lane

__global__ __launch_bounds__(THREADS)
void wordemb_gather_kernel(const float* __restrict__ weight,
                           const int*   __restrict__ ids,
                           float*       __restrict__ out,
                           int n_tokens) {
    const int tid      = threadIdx.x;
    const int base_tok = blockIdx.x * TOKENS_PER_BLOCK;

    if (base_tok + TOKENS_PER_BLOCK <= n_tokens) {
        // ---- Fast path: branchless, single basic block (always taken when
        // n_tokens % TOKENS_PER_BLOCK == 0, as with 16x2048 tokens). ----
        int row[TOKENS_PER_BLOCK];
#pragma unroll
        for (int t = 0; t < TOKENS_PER_BLOCK; ++t)
            row[t] = ids[base_tok + t];            // scalarized s_load clause

        float4v val[TOKENS_PER_BLOCK];
#pragma unroll
        for (int t = 0; t < TOKENS_PER_BLOCK; ++t) {
            const float4v* src = (const float4v*)(weight + (size_t)row[t] * DIM);
            val[t] = src[tid];                     // 4x global_load_b128 in flight
        }

        // Write-once streaming output: NT stores keep the ~200MB weight table
        // (the only reused data) resident in the 192MB L2.
#pragma unroll
        for (int t = 0; t < TOKENS_PER_BLOCK; ++t) {
            float4v* dst = (float4v*)(out + (size_t)(base_tok + t) * DIM);
            __builtin_nontemporal_store(val[t], &dst[tid]);
        }
    } else {
        // ---- Tail path (never taken for 32768 tokens; kept for generality) ----
        for (int t = 0; t < TOKENS_PER_BLOCK; ++t) {
            int tok = base_tok + t;
            if (tok < n_tokens) {
                int r = ids[tok];
                const float4v* src = (const float4v*)(weight + (size_t)r * DIM);
                float4v v = src[tid];
                float4v* dst = (float4v*)(out + (size_t)tok * DIM);
                __builtin_nontemporal_store(v, &dst[tid]);
            }
        }
    }
}

extern "C" void kernel_launch(void* const* d_in, const int* in_sizes, int n_in,
                              void* d_out, int out_size, void* d_ws, size_t ws_size,
                              hipStream_t stream) {
    const float* weight = (const float*)d_in[0];   // [50257*1024] fp32
    const int*   ids    = (const int*)d_in[1];     // [16*2048] int
    float*       out    = (float*)d_out;           // [16*2048*1024] fp32

    const int n_tokens = in_sizes[1];              // 32768
    const int grid     = (n_tokens + TOKENS_PER_BLOCK - 1) / TOKENS_PER_BLOCK;

    wordemb_gather_kernel<<<grid, THREADS, 0, stream>>>(weight, ids, out, n_tokens);
}